// ReceptiveFieldLayer_14680198217840
// MI455X (gfx1250) — compile-verified
//
#include <hip/hip_runtime.h>
#include <hip/hip_bf16.h>
#include <math.h>

// ReceptiveFieldLayer: scatter-max of [16,3,256,256] feature maps (J=4,R=10,pad=3)
// into [16,3,1024,1024] output, then ReLU.  Pure bandwidth problem (~201MB store,
// ~67MB effective load -> ~11.5us roofline at 23.3 TB/s).  No matmul structure, so
// the CDNA5-specific path used here is the async global->LDS copy engine
// (global_load_async_to_lds_b32 + s_wait_asynccnt) staging the feature patch.

#define HF 256
#define WF 256
#define CIN 64
#define HOUT 1024
#define WOUT 1024
#define TILE 32
#define PATCH 10   // feature rows/cols covering a 32-wide output tile: [i0/4-1, i0/4+8]

#if defined(__HIP_DEVICE_COMPILE__) && __has_builtin(__builtin_amdgcn_global_load_async_to_lds_b32) && __has_builtin(__builtin_amdgcn_s_wait_asynccnt)
#define USE_ASYNC_LDS 1
#else
#define USE_ASYNC_LDS 0
#endif

typedef __attribute__((address_space(1))) int global_i32;   // global (AS1) dword
typedef __attribute__((address_space(3))) int local_i32;    // LDS (AS3) dword

__global__ __launch_bounds__(256) void rf_scatter_max_kernel(
    const float* __restrict__ in,   // [16,256,256,64]
    float* __restrict__ out)        // [16,3,1024,1024]
{
    __shared__ float feat[3][PATCH][PATCH];   // 1.2 KB

    const int tid = threadIdx.x;
    const int j0  = blockIdx.x * TILE;        // output tile origin x
    const int i0  = blockIdx.y * TILE;        // output tile origin y
    const int b   = blockIdx.z;
    const int P0  = (i0 >> 2) - 1;            // first feature row in patch
    const int Q0  = (j0 >> 2) - 1;            // first feature col in patch

    // ---------------- stage feature patch (3 channels x 10 x 10) ----------------
    // 300 dwords; invalid (out-of-tensor) slots are never read by the compute phase.
    for (int e = tid; e < 3 * PATCH * PATCH; e += 256) {
        const int t = e / 3;            // feature pixel 0..99
        const int c = e - 3 * t;        // channel 0..2 (lane triplets hit one 64B sector)
        const int r = t / PATCH;
        const int q = t - PATCH * r;
        const int p  = P0 + r;
        const int qg = Q0 + q;
        if (p >= 0 && p < HF && qg >= 0 && qg < WF) {
            float* g = const_cast<float*>(in + ((((size_t)b * HF + p) * WF + qg) * CIN + c));
#if USE_ASYNC_LDS
            __builtin_amdgcn_global_load_async_to_lds_b32(
                (global_i32*)g,
                (local_i32*)&feat[c][r][q],
                /*offset=*/0, /*cpol=*/0);
#else
            feat[c][r][q] = *g;
#endif
        }
    }
#if USE_ASYNC_LDS
    __builtin_amdgcn_s_wait_asynccnt(0);   // each wave drains its own async copies
#endif
    __syncthreads();                       // then all waves' data is visible

    // ---------------- compute 32x32 x 3ch outputs ----------------
    // thread -> one row, 4 consecutive x pixels (float4 store)
    const int ty = tid >> 3;               // 0..31 (row in tile)
    const int x4 = (tid & 7) << 2;         // 0,4,...,28
    const int i  = i0 + ty;                // global output row

    // valid feature-row range for this output row: p in [ceil((i-6)/4), floor((i+3)/4)]
    int p_lo = (i >= 6) ? ((i - 3) >> 2) : 0;
    int p_hi = (i + 3) >> 2; if (p_hi > HF - 1) p_hi = HF - 1;
    const int r_lo = p_lo - P0;            // in [0,9]
    const int r_hi = p_hi - P0;            // in [0,9]

    // the 4 pixels j = 4m+delta all draw from feature cols {m-1, m, m+1}
    const int m  = (j0 + x4) >> 2;
    const int s0 = x4 >> 2;                // patch col index of q = m-1
    const bool c0ok = (m - 1) >= 0;
    const bool c2ok = (m + 1) <= (WF - 1);

    const float NEG = -INFINITY;

    #pragma unroll
    for (int c = 0; c < 3; ++c) {
        float cm0 = NEG, cm1 = NEG, cm2 = NEG;
        for (int r = r_lo; r <= r_hi; ++r) {
            cm0 = fmaxf(cm0, feat[c][r][s0 + 0]);
            cm1 = fmaxf(cm1, feat[c][r][s0 + 1]);
            cm2 = fmaxf(cm2, feat[c][r][s0 + 2]);
        }
        if (!c0ok) cm0 = NEG;              // border cols were never loaded -> mask
        if (!c2ok) cm2 = NEG;

        // delta = 0: {m-1,m}   1,2: {m-1,m,m+1}   3: {m,m+1}   (then ReLU)
        const float m01  = fmaxf(cm0, cm1);
        const float m012 = fmaxf(m01, cm2);
        const float m12  = fmaxf(cm1, cm2);

        float4 v;
        v.x = fmaxf(m01,  0.0f);
        v.y = fmaxf(m012, 0.0f);
        v.z = fmaxf(m012, 0.0f);
        v.w = fmaxf(m12,  0.0f);

        const size_t obase = (((size_t)b * 3 + c) * HOUT + i) * WOUT + (j0 + x4);
        *reinterpret_cast<float4*>(out + obase) = v;
    }
}

extern "C" void kernel_launch(void* const* d_in, const int* in_sizes, int n_in,
                              void* d_out, int out_size, void* d_ws, size_t ws_size,
                              hipStream_t stream) {
    (void)in_sizes; (void)n_in; (void)out_size; (void)d_ws; (void)ws_size;
    const float* in = (const float*)d_in[0];
    float* out = (float*)d_out;

    dim3 grid(WOUT / TILE, HOUT / TILE, 16);   // 32 x 32 x 16 blocks
    dim3 block(256);                            // 8 wave32s
    rf_scatter_max_kernel<<<grid, block, 0, stream>>>(in, out);
}